// SubLSTM_84078279786561
// MI455X (gfx1250) — compile-verified
//
#include <hip/hip_runtime.h>
#include <cstddef>

typedef float v2f __attribute__((ext_vector_type(2)));
typedef float v8f __attribute__((ext_vector_type(8)));
typedef int   v4i_ __attribute__((vector_size(16)));   // pointee type of async-LDS builtin

#define T_STEPS 512
#define BATCH   64
#define HID     512            // IN == H == 512 for both layers
#define GATE    2048
#define KT      32             // K-chunk staged per LDS tile
#define NCHUNK  32             // 16 x-proj + 16 h-proj chunks per step
#define NBLK    64             // persistent blocks (one per 8 features)
#define NTHR    256            // 8 waves (wave32)

// LDS layout (floats), padded strides to avoid bank conflicts
#define WLD     516            // weight row stride (512+4): rows hit distinct banks
#define SLD     36             // staging row stride (32+4): 16B aligned, distinct banks
#define OFF_WI  0
#define OFF_WH  (32 * WLD)
#define OFF_STG (2 * 32 * WLD)
#define STG_BUF (BATCH * SLD)          // one K-chunk buffer
#define OFF_G   (OFF_STG + 3 * STG_BUF)
#define SMEM_FLOATS (OFF_G + BATCH * 32)
#define SMEM_BYTES  (SMEM_FLOATS * 4)

// ---- gfx1250 async global->LDS path (guarded; fallback = sync copy) ----
#if defined(__has_builtin)
#if __has_builtin(__builtin_amdgcn_global_load_async_to_lds_b128)
#define HAVE_ASYNC_LDS 1
#endif
#endif
#ifndef HAVE_ASYNC_LDS
#define HAVE_ASYNC_LDS 0
#endif

#define GAS __attribute__((address_space(1)))
#define LAS __attribute__((address_space(3)))
#define ASYNC_B128(gp, lp) \
    __builtin_amdgcn_global_load_async_to_lds_b128( \
        (GAS v4i_*)(void*)(gp), (LAS v4i_*)(void*)(lp), 0, 0)

#if defined(__has_builtin)
#if __has_builtin(__builtin_amdgcn_s_wait_asynccnt)
#define WAIT_ASYNC(n) __builtin_amdgcn_s_wait_asynccnt(n)
#endif
#endif
#ifndef WAIT_ASYNC
#define WAIT_ASYNC(n) asm volatile("s_wait_asynccnt %0" ::"i"(n) : "memory")
#endif

#define ASYNC_PER_CHUNK 2      // async b128 instructions issued per wave per chunk

// ---- gate-row permutation: dst[(4j+g)*512+k] = src[(g*512+j)*512+k] ----
__global__ void perm_rows_k(const float* __restrict__ src, float* __restrict__ dst) {
    int idx = blockIdx.x * NTHR + threadIdx.x;          // over 2048*512
    int np = idx >> 9;                                  // permuted row
    int k  = idx & 511;
    int j = np >> 2, g = np & 3;
    dst[(size_t)np * 512 + k] = src[((size_t)(g * 512 + j)) * 512 + k];
}

__global__ void perm_bias_k(const float* __restrict__ src, float* __restrict__ dst) {
    int np = blockIdx.x * NTHR + threadIdx.x;           // over 2048
    int j = np >> 2, g = np & 3;
    dst[np] = src[g * 512 + j];
}

__device__ __forceinline__ float sigmoidf_(float v) {
    return 1.0f / (1.0f + __expf(-v));
}

// device-scope generation barrier across the persistent grid
__device__ __forceinline__ void grid_sync(unsigned* cnt, unsigned* gen, int nblk) {
    __threadfence();
    __syncthreads();
    if (threadIdx.x == 0) {
        unsigned g = __hip_atomic_load(gen, __ATOMIC_RELAXED, __HIP_MEMORY_SCOPE_AGENT);
        unsigned a = __hip_atomic_fetch_add(cnt, 1u, __ATOMIC_ACQ_REL, __HIP_MEMORY_SCOPE_AGENT);
        if (a == (unsigned)(nblk - 1)) {
            __hip_atomic_store(cnt, 0u, __ATOMIC_RELAXED, __HIP_MEMORY_SCOPE_AGENT);
            __hip_atomic_fetch_add(gen, 1u, __ATOMIC_RELEASE, __HIP_MEMORY_SCOPE_AGENT);
        } else {
            while (__hip_atomic_load(gen, __ATOMIC_ACQUIRE, __HIP_MEMORY_SCOPE_AGENT) == g)
                __builtin_amdgcn_s_sleep(2);
        }
    }
    __syncthreads();
}

// Persistent subLSTM layer. Block bx owns permuted gate columns [32*bx, 32*bx+32)
// == features [8*bx, 8*bx+8). Wi/Wh slices stay resident in LDS for all T steps;
// cell state c lives in registers; h is exchanged through global + grid barrier.
// Activation K-chunks are triple-buffered in LDS via async global->LDS loads.
__global__ void __launch_bounds__(NTHR)
sublstm_layer(const float* __restrict__ xin,   // [T][64][512]
              const float* __restrict__ Wip,   // [2048][512] permuted
              const float* __restrict__ Whp,   // [2048][512] permuted
              const float* __restrict__ bip,   // [2048] permuted
              float* __restrict__ out,         // [T][64][512]
              float* __restrict__ hbuf,        // [2][64][512], zeroed by host
              unsigned* __restrict__ bar) {    // [2] = {count, gen}, zeroed
    extern __shared__ float smem[];
    float* Wi_l = smem + OFF_WI;
    float* Wh_l = smem + OFF_WH;
    float* stg  = smem + OFF_STG;
    float* G_l  = smem + OFF_G;

    const int tid  = threadIdx.x;
    const int wave = tid >> 5;
    const int lane = tid & 31;
    const int l15  = lane & 15;
    const int hi   = (lane >= 16) ? 2 : 0;   // K sub-offset for this half-wave
    const int mt   = wave >> 1;              // 0..3 -> M rows 16*mt..+16
    const int nt   = wave & 1;               // 0..1 -> N cols 16*nt..+16
    const int n0   = blockIdx.x * 32;        // permuted gate-column base

    // --- load this block's weight slices into LDS (resident for all T) ---
    for (int i = tid; i < 32 * 128; i += NTHR) {      // 4096 float4 per matrix
        int row = i >> 7, c4 = i & 127;
        *(float4*)&Wi_l[row * WLD + c4 * 4] = ((const float4*)(Wip + (size_t)n0 * 512))[i];
        *(float4*)&Wh_l[row * WLD + c4 * 4] = ((const float4*)(Whp + (size_t)n0 * 512))[i];
    }
    const float bias = bip[n0 + nt * 16 + l15];        // per-lane output column bias
    __syncthreads();

    // --- cell state in registers: this thread owns 2 (batch, feature) pairs ---
    const int lin0 = tid * 2, lin1 = tid * 2 + 1;
    const int b0 = lin0 >> 3, f0 = lin0 & 7;
    const int b1 = lin1 >> 3, f1 = lin1 & 7;
    float c0 = 0.0f, c1 = 0.0f;

    float* hcur = hbuf;
    float* hnxt = hbuf + BATCH * HID;

    for (int t = 0; t < T_STEPS; ++t) {
        const float* xt = xin + (size_t)t * BATCH * HID;
        v8f acc = {0.f, 0.f, 0.f, 0.f, 0.f, 0.f, 0.f, 0.f};

#if HAVE_ASYNC_LDS
        // ---- triple-buffered async pipeline over the 32 K-chunks ----
        for (int i = tid; i < (BATCH * KT) / 4; i += NTHR) {   // prologue: chunk 0
            int row = i >> 3, c4 = (i & 7) * 4;
            ASYNC_B128(xt + (size_t)row * HID + c4, stg + row * SLD + c4);
        }
        for (int q = 0; q < NCHUNK; ++q) {
            if (q + 1 < NCHUNK) {
                const int qn = q + 1;
                const float* An = (qn < 16) ? xt : hcur;
                const int k0n = (qn & 15) * KT;
                float* dn = stg + (qn % 3) * STG_BUF;
                for (int i = tid; i < (BATCH * KT) / 4; i += NTHR) {
                    int row = i >> 3, c4 = (i & 7) * 4;
                    ASYNC_B128(An + (size_t)row * HID + k0n + c4, dn + row * SLD + c4);
                }
                WAIT_ASYNC(ASYNC_PER_CHUNK);   // chunk q landed; q+1 in flight
            } else {
                WAIT_ASYNC(0);
            }
            __syncthreads();                   // all waves see chunk q in LDS

            const float* sb = stg + (q % 3) * STG_BUF;
            const float* Wl = (q < 16) ? Wi_l : Wh_l;
            const int k0 = (q & 15) * KT;
            #pragma unroll
            for (int kk = 0; kk < KT; kk += 4) {
                const int ar = (mt * 16 + l15) * SLD + kk + hi;
                v2f af; af.x = sb[ar]; af.y = sb[ar + 1];
                const int br = (nt * 16 + l15) * WLD + k0 + kk + hi;
                v2f bf; bf.x = Wl[br]; bf.y = Wl[br + 1];
                acc = __builtin_amdgcn_wmma_f32_16x16x4_f32(
                    false, af, false, bf, (short)0, acc, false, false);
            }
        }
#else
        // ---- fallback: synchronous staging (two barriers per chunk) ----
        for (int q = 0; q < NCHUNK; ++q) {
            const float* A  = (q < 16) ? xt : hcur;
            const float* Wl = (q < 16) ? Wi_l : Wh_l;
            const int k0 = (q & 15) * KT;
            float* sb = stg;                   // single buffer
            __syncthreads();
            for (int i = tid; i < (BATCH * KT) / 2; i += NTHR) {
                int row = i >> 4, c2 = (i & 15) * 2;
                *(float2*)&sb[row * SLD + c2] =
                    *(const float2*)(A + (size_t)row * HID + k0 + c2);
            }
            __syncthreads();
            #pragma unroll
            for (int kk = 0; kk < KT; kk += 4) {
                const int ar = (mt * 16 + l15) * SLD + kk + hi;
                v2f af; af.x = sb[ar]; af.y = sb[ar + 1];
                const int br = (nt * 16 + l15) * WLD + k0 + kk + hi;
                v2f bf; bf.x = Wl[br]; bf.y = Wl[br + 1];
                acc = __builtin_amdgcn_wmma_f32_16x16x4_f32(
                    false, af, false, bf, (short)0, acc, false, false);
            }
        }
#endif

        // --- bias + sigmoid, scatter tile to LDS for the gate combine ---
        #pragma unroll
        for (int r = 0; r < 8; ++r) {
            const int m = mt * 16 + r + (hi ? 8 : 0);   // C/D layout: hi-half -> M+8
            const int n = nt * 16 + l15;
            G_l[m * 32 + n] = sigmoidf_(acc[r] + bias);
        }
        __syncthreads();

        // --- subtractive cell update: cols 4f+{0,1,2,3} = {in, out, z, forget} ---
        {
            float4 g0 = *(const float4*)&G_l[b0 * 32 + f0 * 4];
            c0 = c0 * g0.w + g0.z - g0.x;
            float h0 = sigmoidf_(c0) - g0.y;
            float4 g1 = *(const float4*)&G_l[b1 * 32 + f1 * 4];
            c1 = c1 * g1.w + g1.z - g1.x;
            float h1 = sigmoidf_(c1) - g1.y;

            const int j0 = blockIdx.x * 8 + f0;
            const int j1 = blockIdx.x * 8 + f1;
            hnxt[b0 * HID + j0] = h0;
            hnxt[b1 * HID + j1] = h1;
            out[((size_t)t * BATCH + b0) * HID + j0] = h0;
            out[((size_t)t * BATCH + b1) * HID + j1] = h1;
        }

        grid_sync(bar, bar + 1, NBLK);
        float* tmp = hcur; hcur = hnxt; hnxt = tmp;
    }
}

extern "C" void kernel_launch(void* const* d_in, const int* in_sizes, int n_in,
                              void* d_out, int out_size, void* d_ws, size_t ws_size,
                              hipStream_t stream) {
    const float* x   = (const float*)d_in[0];
    const float* Wi0 = (const float*)d_in[1];
    const float* bi0 = (const float*)d_in[2];
    const float* Wh0 = (const float*)d_in[3];
    const float* Wi1 = (const float*)d_in[4];
    const float* bi1 = (const float*)d_in[5];
    const float* Wh1 = (const float*)d_in[6];
    float* out = (float*)d_out;

    char* ws = (char*)d_ws;
    size_t o = 0;
    const size_t WMAT = (size_t)GATE * HID * sizeof(float);   // 4 MB
    float* Wi0p = (float*)(ws + o); o += WMAT;
    float* Wh0p = (float*)(ws + o); o += WMAT;
    float* Wi1p = (float*)(ws + o); o += WMAT;
    float* Wh1p = (float*)(ws + o); o += WMAT;
    float* bi0p = (float*)(ws + o); o += GATE * sizeof(float);
    float* bi1p = (float*)(ws + o); o += GATE * sizeof(float);
    float* hbuf = (float*)(ws + o); o += 2ull * BATCH * HID * sizeof(float);
    unsigned* bar = (unsigned*)(ws + o); o += 256;            // keep alignment
    float* l0out = (float*)(ws + o);                          // 64 MB

    const int PERM_BLKS = (GATE * HID) / NTHR;                // 4096

    perm_rows_k<<<PERM_BLKS, NTHR, 0, stream>>>(Wi0, Wi0p);
    perm_rows_k<<<PERM_BLKS, NTHR, 0, stream>>>(Wh0, Wh0p);
    perm_rows_k<<<PERM_BLKS, NTHR, 0, stream>>>(Wi1, Wi1p);
    perm_rows_k<<<PERM_BLKS, NTHR, 0, stream>>>(Wh1, Wh1p);
    perm_bias_k<<<GATE / NTHR, NTHR, 0, stream>>>(bi0, bi0p);
    perm_bias_k<<<GATE / NTHR, NTHR, 0, stream>>>(bi1, bi1p);

    (void)hipMemsetAsync(bar, 0, 2 * sizeof(unsigned), stream);
    (void)hipMemsetAsync(hbuf, 0, 2ull * BATCH * HID * sizeof(float), stream);
    sublstm_layer<<<NBLK, NTHR, SMEM_BYTES, stream>>>(
        x, Wi0p, Wh0p, bi0p, l0out, hbuf, bar);

    (void)hipMemsetAsync(hbuf, 0, 2ull * BATCH * HID * sizeof(float), stream);
    sublstm_layer<<<NBLK, NTHR, SMEM_BYTES, stream>>>(
        l0out, Wi1p, Wh1p, bi1p, out, hbuf, bar);
}